// SAGEHeteroConv_40261023432832
// MI455X (gfx1250) — compile-verified
//
#include <hip/hip_runtime.h>

// SAGE hetero conv for MI455X (gfx1250, wave32, WMMA).
//
// out_item = x_item @ W_ui_tgt + scatter_mean(x_user @ W_ui_src, edge_ui)
// out_user = x_user @ W_iu_tgt + scatter_mean(x_item @ W_iu_src, edge_iu)
//
// GEMMs use V_WMMA_F32_16X16X4_F32 (full f32 precision; compute is ~10 GFLOP,
// nowhere near the bottleneck -- the kernel is L2-atomic / HBM bound).

typedef float v2f __attribute__((ext_vector_type(2)));
typedef float v4f __attribute__((ext_vector_type(4)));
typedef float v8f __attribute__((ext_vector_type(8)));

#define DIM  128   // input feature dim (K)
#define OUTF 128   // output feature dim (N)

// ---------------------------------------------------------------------------
// Zero-fill
// ---------------------------------------------------------------------------
__global__ void sage_zero_kernel(float* __restrict__ p, long n) {
  long i = (long)blockIdx.x * blockDim.x + threadIdx.x;
  long stride = (long)gridDim.x * blockDim.x;
  for (; i < n; i += stride) p[i] = 0.0f;
}

// ---------------------------------------------------------------------------
// WMMA f32 GEMM: Y = X[nrows x 128] @ W[128 x 128]
//   mode 0: Y = X@W
//   mode 1: Y = X@W + Y_in / max(cnt,1)   (fused mean + residual finalize)
//
// One wave (32 lanes) computes a 16-row x 128-col slab of Y:
//   8 accumulator tiles (16x16 f32, v8f each), K swept in steps of 4 with
//   V_WMMA_F32_16X16X4_F32.
//
// f32 WMMA operand layouts (ISA 7.12.2):
//   A (16x4, MxK):  lane L holds M = L%16; VGPR j holds K = 2*(L/16) + j
//   B (4x16, KxN):  lane L holds N = L%16; VGPR j holds K = 2*(L/16) + j
//   C/D (16x16):    lane L, VGPR r -> row M = r + 8*(L/16), col N = L%16
// ---------------------------------------------------------------------------
__global__ void sage_gemm_kernel(const float* __restrict__ X,
                                 const float* __restrict__ W,
                                 float* __restrict__ Y,
                                 const float* __restrict__ cnt,
                                 int nrows, int mode) {
  const int wave = threadIdx.x >> 5;
  const int lane = threadIdx.x & 31;
  const int tile = blockIdx.x * (blockDim.x >> 5) + wave;  // 16-row tile id
  const int row0 = tile * 16;
  if (row0 >= nrows) return;  // wave-uniform: EXEC stays all-ones below

  const int m    = lane & 15;        // row within A tile / col within B tile
  const int half = lane >> 4;        // 0 or 1
  const int kh   = half << 1;        // K sub-offset: 0 or 2

  // clamp row for loads (all lanes stay active -> EXEC all-ones for WMMA)
  int arow = row0 + m;
  if (arow > nrows - 1) arow = nrows - 1;
  const float* __restrict__ xrow = X + (long)arow * DIM;

  v8f acc[8] = {};

  #pragma unroll 4
  for (int k = 0; k < DIM; k += 4) {
    const v2f a = *(const v2f*)(xrow + k + kh);          // global_load_b64
    #pragma unroll
    for (int t = 0; t < 8; ++t) {
      const int col = t * 16 + m;
      v2f b;
      b.x = W[(long)(k + kh)     * OUTF + col];
      b.y = W[(long)(k + kh + 1) * OUTF + col];
      acc[t] = __builtin_amdgcn_wmma_f32_16x16x4_f32(
          /*neg_a=*/false, a, /*neg_b=*/false, b,
          /*c_mod=*/(short)0, acc[t],
          /*reuse_a=*/false, /*reuse_b=*/false);
    }
  }

  // Write back per the C/D layout.
  #pragma unroll
  for (int r = 0; r < 8; ++r) {
    const int row = row0 + r + 8 * half;
    if (row < nrows) {
      const float inv = (mode == 1) ? (1.0f / fmaxf(cnt[row], 1.0f)) : 0.0f;
      #pragma unroll
      for (int t = 0; t < 8; ++t) {
        const long idx = (long)row * OUTF + t * 16 + (lane & 15);
        float v = acc[t][r];
        if (mode == 1) v += Y[idx] * inv;   // sum/max(cnt,1) + tgt projection
        Y[idx] = v;
      }
    }
  }
}

// ---------------------------------------------------------------------------
// Edge scatter: sum[tgt] += proj[src] (128 floats), cnt[tgt] += 1.
// 32 lanes cooperate per edge; each lane moves a float4 (b128 gather +
// 4x global_atomic_add_f32).
// ---------------------------------------------------------------------------
__global__ void sage_scatter_kernel(const float* __restrict__ proj,
                                    const int* __restrict__ esrc,
                                    const int* __restrict__ etgt,
                                    float* __restrict__ sum,
                                    float* __restrict__ cnt,
                                    int nedges) {
  long tid = (long)blockIdx.x * blockDim.x + threadIdx.x;
  const long total  = (long)nedges * 32;
  const long stride = (long)gridDim.x * blockDim.x;
  for (; tid < total; tid += stride) {
    const int e = (int)(tid >> 5);
    const int l = (int)(tid & 31);
    const int s = esrc[e];
    const int t = etgt[e];
    const v4f v = *(const v4f*)(proj + (long)s * OUTF + l * 4);  // b128 gather
    float* __restrict__ dst = sum + (long)t * OUTF + l * 4;
    atomicAdd(dst + 0, v.x);
    atomicAdd(dst + 1, v.y);
    atomicAdd(dst + 2, v.z);
    atomicAdd(dst + 3, v.w);
    if (l == 0) atomicAdd(cnt + t, 1.0f);
  }
}

// ---------------------------------------------------------------------------
// Launch
// ---------------------------------------------------------------------------
extern "C" void kernel_launch(void* const* d_in, const int* in_sizes, int n_in,
                              void* d_out, int out_size, void* d_ws, size_t ws_size,
                              hipStream_t stream) {
  const float* x_user   = (const float*)d_in[0];
  const float* x_item   = (const float*)d_in[1];
  const float* W_ui_src = (const float*)d_in[2];
  const float* W_ui_tgt = (const float*)d_in[3];
  const float* W_iu_src = (const float*)d_in[4];
  const float* W_iu_tgt = (const float*)d_in[5];
  const int*   edge_ui  = (const int*)d_in[6];   // [2, E]: row0 = user src, row1 = item tgt
  const int*   edge_iu  = (const int*)d_in[7];   // [2, E]: row0 = item src, row1 = user tgt

  const int n_user = in_sizes[0] / DIM;   // 100000
  const int n_item = in_sizes[1] / DIM;   // 50000
  const int E      = in_sizes[6] / 2;     // 1600000

  float* out_user = (float*)d_out;                         // [n_user, 128]
  float* out_item = (float*)d_out + (long)n_user * OUTF;   // [n_item, 128]

  // workspace layout (floats): proj_u | proj_i | cnt_item | cnt_user
  float* proj_u   = (float*)d_ws;
  float* proj_i   = proj_u + (long)n_user * OUTF;
  float* cnt_item = proj_i + (long)n_item * OUTF;
  float* cnt_user = cnt_item + n_item;

  // 1) zero the sums (d_out) and counts
  const long nout = (long)(n_user + n_item) * OUTF;
  sage_zero_kernel<<<2048, 256, 0, stream>>>((float*)d_out, nout);
  sage_zero_kernel<<<128, 256, 0, stream>>>(cnt_item, (long)(n_item + n_user));

  // 2) source projections (WMMA GEMM) into workspace
  const int tiles_u  = (n_user + 15) / 16;
  const int tiles_i  = (n_item + 15) / 16;
  const int blocks_u = (tiles_u + 7) / 8;   // 8 waves / 256-thread block
  const int blocks_i = (tiles_i + 7) / 8;
  sage_gemm_kernel<<<blocks_u, 256, 0, stream>>>(x_user, W_ui_src, proj_u,
                                                 nullptr, n_user, 0);
  sage_gemm_kernel<<<blocks_i, 256, 0, stream>>>(x_item, W_iu_src, proj_i,
                                                 nullptr, n_item, 0);

  // 3) scatter-add projected source features + counts
  sage_scatter_kernel<<<16384, 256, 0, stream>>>(proj_u, edge_ui, edge_ui + E,
                                                 out_item, cnt_item, E);
  sage_scatter_kernel<<<16384, 256, 0, stream>>>(proj_i, edge_iu, edge_iu + E,
                                                 out_user, cnt_user, E);

  // 4) finalize: out = x_tgt @ W_tgt + sum / max(cnt, 1)   (fused WMMA GEMM)
  sage_gemm_kernel<<<blocks_i, 256, 0, stream>>>(x_item, W_ui_tgt, out_item,
                                                 cnt_item, n_item, 1);
  sage_gemm_kernel<<<blocks_u, 256, 0, stream>>>(x_user, W_iu_tgt, out_user,
                                                 cnt_user, n_user, 1);
}